// Model_80152679678605
// MI455X (gfx1250) — compile-verified
//
#include <hip/hip_runtime.h>

typedef __attribute__((ext_vector_type(2))) float v2f;
typedef __attribute__((ext_vector_type(8))) float v8f;
typedef __attribute__((ext_vector_type(4))) unsigned int u32x4;
typedef __attribute__((ext_vector_type(8))) int i32x8;
typedef __attribute__((ext_vector_type(4))) int i32x4;

#define ATOMIC_FADD(p, v) \
  __hip_atomic_fetch_add((p), (v), __ATOMIC_RELAXED, __HIP_MEMORY_SCOPE_AGENT)

#if defined(__gfx1250__) && __has_builtin(__builtin_amdgcn_tensor_load_to_lds)
#define SAGE_TDM 1
#else
#define SAGE_TDM 0
#endif

// x_s[i] = src_emb[ids[i]]  (row gather, float4 granularity, H=64 -> 16 chunks)
__global__ void k_gather_rows(const float4* __restrict__ emb,
                              const int* __restrict__ ids,
                              float4* __restrict__ out, int n) {
  int tid = blockIdx.x * blockDim.x + threadIdx.x;
  if (tid >= n * 16) return;
  int i = tid >> 4, c = tid & 15;
  out[(size_t)i * 16 + c] = emb[(size_t)ids[i] * 16 + c];
}

// x_t = target_x @ lin_W + lin_b + tgt_emb[ids]   (K=20, tiny GEMM, scalar is fine)
__global__ void k_xt(const float* __restrict__ tx, const float* __restrict__ W,
                     const float* __restrict__ b, const float* __restrict__ emb,
                     const int* __restrict__ ids, float* __restrict__ out,
                     int nt, int df) {
  int tid = blockIdx.x * blockDim.x + threadIdx.x;
  if (tid >= nt * 64) return;
  int i = tid >> 6, j = tid & 63;
  float acc = b[j];
  const float* xr = tx + (size_t)i * df;
  for (int k = 0; k < df; ++k) acc = fmaf(xr[k], W[(size_t)k * 64 + j], acc);
  out[tid] = acc + emb[(size_t)ids[i] * 64 + j];
}

// degree of both endpoints in one pass
__global__ void k_deg(const int* __restrict__ es, const int* __restrict__ ed,
                      float* __restrict__ degs, float* __restrict__ degt, int ne) {
  int e = blockIdx.x * blockDim.x + threadIdx.x;
  if (e >= ne) return;
  ATOMIC_FADD(&degs[es[e]], 1.0f);
  ATOMIC_FADD(&degt[ed[e]], 1.0f);
}

// agg[di[e]] += x[si[e]]   (16 threads/edge, float4 per thread, native f32 atomics)
__global__ void k_scatter(const float4* __restrict__ x, const int* __restrict__ si,
                          const int* __restrict__ di, float* __restrict__ agg, int ne) {
  long tid = (long)blockIdx.x * blockDim.x + threadIdx.x;
  if (tid >= (long)ne * 16) return;
  int e = (int)(tid >> 4), c = (int)(tid & 15);
  float4 v = x[(size_t)si[e] * 16 + c];
  float* dst = agg + (size_t)di[e] * 64 + (size_t)c * 4;
  ATOMIC_FADD(dst + 0, v.x);
  ATOMIC_FADD(dst + 1, v.y);
  ATOMIC_FADD(dst + 2, v.z);
  ATOMIC_FADD(dst + 3, v.w);
}

#if SAGE_TDM
// Issue one TDM descriptor: 64x64 f32 tile, row-major, global -> LDS.
// D# built per CDNA5 ISA ch.8: group0 {count=1, lds_addr, global_addr, type=2},
// group1 {data_size=4B, tensor 64x64 stride 64, tile 64x64}.
// This toolchain's builtin is the 6-arg clang-23 form:
//   (u32x4 g0, i32x8 g1, i32x4 g2, i32x4 g3, i32x8 extra, i32 cpol)
__device__ __forceinline__ void tdm_load_w64(unsigned lds_off, const float* gptr) {
  unsigned long long ga = (unsigned long long)(uintptr_t)gptr;
  u32x4 g0;
  g0[0] = 1u;                                        // count=1 (valid user D#)
  g0[1] = lds_off;                                   // lds_addr (bytes)
  g0[2] = (unsigned)(ga & 0xFFFFFFFFu);              // global_addr[31:0]
  g0[3] = (unsigned)((ga >> 32) & 0x1FFFFFFu)        // global_addr[56:32]
          | (2u << 30);                              // type = 2 ("image")
  i32x8 g1;
  g1[0] = (int)(2u << 16);                           // data_size=2 -> 4 bytes
  g1[1] = (int)(64u << 16);                          // tensor_dim0[15:0] = 64
  g1[2] = (int)(64u << 16);                          // dim0 hi=0 | tensor_dim1[15:0]=64
  g1[3] = (int)(64u << 16);                          // dim1 hi=0 | tile_dim0=64
  g1[4] = 64;                                        // tile_dim1=64, tile_dim2=0
  g1[5] = 64;                                        // tensor_dim0_stride[31:0]=64
  g1[6] = (int)(4096u << 16);                        // dim0_stride hi=0 | dim1_stride lo
  g1[7] = 0;                                         // dim1_stride hi
  i32x4 gz4 = {0, 0, 0, 0};                          // groups 2/3 unused (2D)
  i32x8 gz8 = {0, 0, 0, 0, 0, 0, 0, 0};              // extra group (unused)
  __builtin_amdgcn_tensor_load_to_lds(g0, g1, gz4, gz4, gz8, 0);
}
#endif

// out = relu?( (agg/max(deg,1)) @ Wl + bl + xd @ Wr )
// Fused K=128 GEMM [mean | xd] @ [[Wl];[Wr]]. Weights staged in LDS (TDM if
// available, cooperative copy otherwise); one wave per 16-row strip, 4 N-tiles,
// 32 k-steps of V_WMMA_F32_16X16X4_F32 => 128 WMMAs/wave, FP32-exact.
__global__ void __launch_bounds__(256)
k_sage(const float* __restrict__ agg, const float* __restrict__ deg,
       const float* __restrict__ xd, const float* __restrict__ Wl,
       const float* __restrict__ bl, const float* __restrict__ Wr,
       float* __restrict__ out, int nrows, int relu) {
  __shared__ float lw[128 * 64];                     // fused [Wl;Wr], 32KB

#if SAGE_TDM
  if (threadIdx.x < 32) {                            // wave 0 drives the TDM
    unsigned lbase = (unsigned)(uintptr_t)(&lw[0]);  // low 32b of flat = LDS offset
    tdm_load_w64(lbase, Wl);
    tdm_load_w64(lbase + 64u * 64u * 4u, Wr);
    __builtin_amdgcn_s_wait_tensorcnt(0);
  }
#else
  for (int i = threadIdx.x; i < 2048; i += (int)blockDim.x) {
    float4 v = (i < 1024) ? ((const float4*)Wl)[i] : ((const float4*)Wr)[i - 1024];
    ((float4*)lw)[i] = v;
  }
#endif
  __syncthreads();

  int wave = (blockIdx.x * blockDim.x + threadIdx.x) >> 5;
  int lane = threadIdx.x & 31;
  int base = wave * 16;
  if (base < nrows) {                  // wave-uniform: EXEC all-ones for WMMA
    const int lo  = lane & 15;
    const int hi  = lane >> 4;
    const int hi2 = hi * 2;

    int row = base + lo;
    if (row >= nrows) row = nrows - 1;               // load clamp
    const float rinv = 1.0f / fmaxf(deg[row], 1.0f);
    const float* arow = agg + (size_t)row * 64;
    const float* xrow = xd  + (size_t)row * 64;

    v8f acc0 = {}, acc1 = {}, acc2 = {}, acc3 = {};

    // B fragment from LDS: b[v] = W[k0 + v][ nb*16 + lo ], k0 = 4*kk + 2*hi
    auto ldb = [&](int k0, int nb) -> v2f {
      v2f b;
      b.x = lw[k0 * 64 + nb * 16 + lo];
      b.y = lw[(k0 + 1) * 64 + nb * 16 + lo];
      return b;
    };

#pragma unroll
    for (int kk = 0; kk < 32; ++kk) {
      const int k0 = kk * 4 + hi2;                   // fused K index 0..127
      v2f a;
      if (kk < 16) {                                 // mean part -> Wl rows of lw
        a.x = arow[k0] * rinv;
        a.y = arow[k0 + 1] * rinv;
      } else {                                       // x_dst part -> Wr rows of lw
        a.x = xrow[k0 - 64];
        a.y = xrow[k0 - 63];
      }
      v2f b0 = ldb(k0, 0), b1 = ldb(k0, 1), b2 = ldb(k0, 2), b3 = ldb(k0, 3);
      acc0 = __builtin_amdgcn_wmma_f32_16x16x4_f32(false, a, false, b0, (short)0, acc0, false, false);
      acc1 = __builtin_amdgcn_wmma_f32_16x16x4_f32(false, a, false, b1, (short)0, acc1, false, false);
      acc2 = __builtin_amdgcn_wmma_f32_16x16x4_f32(false, a, false, b2, (short)0, acc2, false, false);
      acc3 = __builtin_amdgcn_wmma_f32_16x16x4_f32(false, a, false, b3, (short)0, acc3, false, false);
    }

    const float cb0 = bl[lo], cb1 = bl[16 + lo], cb2 = bl[32 + lo], cb3 = bl[48 + lo];
#pragma unroll
    for (int v = 0; v < 8; ++v) {
      int r = base + v + hi * 8;                     // D layout: VGPR v -> row v + 8*hi
      if (r < nrows) {
        size_t ro = (size_t)r * 64 + lo;
        float o0 = acc0[v] + cb0, o1 = acc1[v] + cb1;
        float o2 = acc2[v] + cb2, o3 = acc3[v] + cb3;
        if (relu) {
          o0 = fmaxf(o0, 0.f); o1 = fmaxf(o1, 0.f);
          o2 = fmaxf(o2, 0.f); o3 = fmaxf(o3, 0.f);
        }
        out[ro] = o0; out[ro + 16] = o1; out[ro + 32] = o2; out[ro + 48] = o3;
      }
    }
  }
}

// out[l] = dot(o_s[ls[l]], o_t[ld[l]])  — both operand tables are L2-resident
__global__ void k_dot(const float4* __restrict__ os, const float4* __restrict__ ot,
                      const int* __restrict__ ls, const int* __restrict__ ld,
                      float* __restrict__ out, int nl) {
  int l = blockIdx.x * blockDim.x + threadIdx.x;
  if (l >= nl) return;
  const float4* a = os + (size_t)ls[l] * 16;
  const float4* b = ot + (size_t)ld[l] * 16;
  float acc = 0.f;
#pragma unroll
  for (int k = 0; k < 16; ++k) {
    float4 x = a[k], y = b[k];
    acc += x.x * y.x + x.y * y.y + x.z * y.z + x.w * y.w;
  }
  out[l] = acc;
}

extern "C" void kernel_launch(void* const* d_in, const int* in_sizes, int n_in,
                              void* d_out, int out_size, void* d_ws, size_t ws_size,
                              hipStream_t stream) {
  (void)n_in; (void)out_size; (void)ws_size;
  const float* target_x = (const float*)d_in[0];
  const int*   src_nid  = (const int*)d_in[1];
  const int*   tgt_nid  = (const int*)d_in[2];
  const int*   edge_src = (const int*)d_in[3];
  const int*   edge_dst = (const int*)d_in[4];
  const int*   lbl_src  = (const int*)d_in[5];
  const int*   lbl_dst  = (const int*)d_in[6];
  const float* src_emb  = (const float*)d_in[7];
  const float* tgt_emb  = (const float*)d_in[8];
  const float* lin_W    = (const float*)d_in[9];
  const float* lin_b    = (const float*)d_in[10];
  const float *Wl_st1=(const float*)d_in[11], *bl_st1=(const float*)d_in[12], *Wr_st1=(const float*)d_in[13];
  const float *Wl_ts1=(const float*)d_in[14], *bl_ts1=(const float*)d_in[15], *Wr_ts1=(const float*)d_in[16];
  const float *Wl_st2=(const float*)d_in[17], *bl_st2=(const float*)d_in[18], *Wr_st2=(const float*)d_in[19];
  const float *Wl_ts2=(const float*)d_in[20], *bl_ts2=(const float*)d_in[21], *Wr_ts2=(const float*)d_in[22];

  const int NS = in_sizes[1];
  const int NT = in_sizes[2];
  const int NE = in_sizes[3];
  const int NL = in_sizes[5];
  const int DF = in_sizes[0] / NT;

  // workspace partition (floats). xs/xt are reused as o_s/o_t in layer 2.
  float* p    = (float*)d_ws;
  float* xs   = p; p += (size_t)NS * 64;   // x_s, later o_s
  float* xt   = p; p += (size_t)NT * 64;   // x_t, later o_t
  float* hs   = p; p += (size_t)NS * 64;
  float* ht   = p; p += (size_t)NT * 64;
  float* aggs = p; p += (size_t)NS * 64;   // agg_s then agg_t, contiguous
  float* aggt = p; p += (size_t)NT * 64;
  float* degs = p; p += (size_t)NS;        // deg_s then deg_t, contiguous
  float* degt = p; p += (size_t)NT;

  const int B = 256;
  const size_t aggBytes = ((size_t)NS + NT) * 64 * sizeof(float);
  (void)hipMemsetAsync(degs, 0, ((size_t)NS + NT) * sizeof(float), stream);
  (void)hipMemsetAsync(aggs, 0, aggBytes, stream);

  k_gather_rows<<<(NS * 16 + B - 1) / B, B, 0, stream>>>((const float4*)src_emb, src_nid, (float4*)xs, NS);
  k_xt<<<(NT * 64 + B - 1) / B, B, 0, stream>>>(target_x, lin_W, lin_b, tgt_emb, tgt_nid, xt, NT, DF);
  k_deg<<<(NE + B - 1) / B, B, 0, stream>>>(edge_src, edge_dst, degs, degt, NE);

  const int scatterBlocks = (int)(((long)NE * 16 + B - 1) / B);
  const int gT = ((NT + 15) / 16 + 7) / 8;   // waves of 16 rows, 8 waves/block
  const int gS = ((NS + 15) / 16 + 7) / 8;

  // ---- layer 1 ----
  k_scatter<<<scatterBlocks, B, 0, stream>>>((const float4*)xs, edge_src, edge_dst, aggt, NE);
  k_sage<<<gT, B, 0, stream>>>(aggt, degt, xt, Wl_st1, bl_st1, Wr_st1, ht, NT, 1);
  k_scatter<<<scatterBlocks, B, 0, stream>>>((const float4*)xt, edge_dst, edge_src, aggs, NE);
  k_sage<<<gS, B, 0, stream>>>(aggs, degs, xs, Wl_ts1, bl_ts1, Wr_ts1, hs, NS, 1);

  // ---- layer 2 ----
  (void)hipMemsetAsync(aggs, 0, aggBytes, stream);
  k_scatter<<<scatterBlocks, B, 0, stream>>>((const float4*)hs, edge_src, edge_dst, aggt, NE);
  k_sage<<<gT, B, 0, stream>>>(aggt, degt, ht, Wl_st2, bl_st2, Wr_st2, xt /*o_t*/, NT, 0);
  k_scatter<<<scatterBlocks, B, 0, stream>>>((const float4*)ht, edge_dst, edge_src, aggs, NE);
  k_sage<<<gS, B, 0, stream>>>(aggs, degs, hs, Wl_ts2, bl_ts2, Wr_ts2, xs /*o_s*/, NS, 0);

  // ---- link classifier ----
  k_dot<<<(NL + B - 1) / B, B, 0, stream>>>((const float4*)xs, (const float4*)xt,
                                            lbl_src, lbl_dst, (float*)d_out, NL);
}